// SageRes_13202729468518
// MI455X (gfx1250) — compile-verified
//
#include <hip/hip_runtime.h>
#include <hip/hip_bf16.h>

#define NN 100000
#define NE 3200000

typedef __attribute__((ext_vector_type(2))) float v2f;
typedef __attribute__((ext_vector_type(8))) float v8f;

// ---------- CDNA5 async global->LDS copy (ASYNCcnt-tracked), inline asm ----------
__device__ __forceinline__ void async_copy_b128(unsigned lds_addr, const void* gptr) {
  asm volatile("global_load_async_to_lds_b128 %0, %1, off"
               :: "v"(lds_addr), "v"((unsigned long long)(uintptr_t)gptr)
               : "memory");
}
__device__ __forceinline__ void wait_asynccnt0() {
  asm volatile("s_wait_asynccnt 0x0" ::: "memory");
}

// ---------- float atomic min/max via integer ordering trick ----------
__device__ __forceinline__ void atomicMaxF(float* addr, float v) {
  if (v >= 0.0f) atomicMax((int*)addr, __float_as_int(v));
  else           atomicMin((unsigned int*)addr, __float_as_uint(v));
}
__device__ __forceinline__ void atomicMinF(float* addr, float v) {
  if (v >= 0.0f) atomicMin((int*)addr, __float_as_int(v));
  else           atomicMax((unsigned int*)addr, __float_as_uint(v));
}

// ---------- utility ----------
__global__ void zero_kernel(float* __restrict__ p, long long n) {
  long long i = (long long)blockIdx.x * blockDim.x + threadIdx.x;
  long long stride = (long long)gridDim.x * blockDim.x;
  for (; i < n; i += stride) p[i] = 0.0f;
}

// ---------- normalization ----------
// stats layout: 0:max_x 1:max_y 2:min_x 3:min_y 4:sum_sx 5:sum_sy 6:max_sx 7:max_sy 8:max_area
__global__ void stats_init_kernel(float* __restrict__ st) {
  if (threadIdx.x == 0) {
    st[0] = -3.402823466e38f; st[1] = -3.402823466e38f;
    st[2] =  3.402823466e38f; st[3] =  3.402823466e38f;
    st[4] = 0.0f; st[5] = 0.0f;
    st[6] = -3.402823466e38f; st[7] = -3.402823466e38f;
    st[8] = -3.402823466e38f;
  }
}

__global__ void reduce_minmax_kernel(const float* __restrict__ x, float* __restrict__ st) {
  long long i = (long long)blockIdx.x * blockDim.x + threadIdx.x;
  long long stride = (long long)gridDim.x * blockDim.x;
  float mx0 = -3.402823466e38f, mx1 = -3.402823466e38f;
  float mn0 =  3.402823466e38f, mn1 =  3.402823466e38f;
  for (; i < NN; i += stride) {
    float a = x[i * 3 + 0], b = x[i * 3 + 1];
    mx0 = fmaxf(mx0, a); mx1 = fmaxf(mx1, b);
    mn0 = fminf(mn0, a); mn1 = fminf(mn1, b);
  }
  for (int off = 16; off > 0; off >>= 1) {
    mx0 = fmaxf(mx0, __shfl_xor(mx0, off, 32));
    mx1 = fmaxf(mx1, __shfl_xor(mx1, off, 32));
    mn0 = fminf(mn0, __shfl_xor(mn0, off, 32));
    mn1 = fminf(mn1, __shfl_xor(mn1, off, 32));
  }
  if ((threadIdx.x & 31) == 0) {
    atomicMaxF(&st[0], mx0); atomicMaxF(&st[1], mx1);
    atomicMinF(&st[2], mn0); atomicMinF(&st[3], mn1);
  }
}

__global__ void reduce_stats2_kernel(const float* __restrict__ x, float* __restrict__ st) {
  bool cond = (st[1] - st[3]) > (st[0] - st[2]);
  const float th = 1.5707963267948966f;
  float cth = cosf(th), sth = sinf(th);
  long long i = (long long)blockIdx.x * blockDim.x + threadIdx.x;
  long long stride = (long long)gridDim.x * blockDim.x;
  float s0 = 0.0f, s1 = 0.0f;
  float mx0 = -3.402823466e38f, mx1 = -3.402823466e38f, mxa = -3.402823466e38f;
  for (; i < NN; i += stride) {
    float cx = x[i * 3 + 0], cy = x[i * 3 + 1], ar = x[i * 3 + 2];
    float r0 = cx * cth - cy * sth;
    float r1 = cx * sth + cy * cth;
    float a = cond ? r0 : cx;
    float b = cond ? r1 : cy;
    s0 += a; s1 += b;
    mx0 = fmaxf(mx0, a); mx1 = fmaxf(mx1, b); mxa = fmaxf(mxa, ar);
  }
  for (int off = 16; off > 0; off >>= 1) {
    s0 += __shfl_xor(s0, off, 32);
    s1 += __shfl_xor(s1, off, 32);
    mx0 = fmaxf(mx0, __shfl_xor(mx0, off, 32));
    mx1 = fmaxf(mx1, __shfl_xor(mx1, off, 32));
    mxa = fmaxf(mxa, __shfl_xor(mxa, off, 32));
  }
  if ((threadIdx.x & 31) == 0) {
    atomicAdd(&st[4], s0); atomicAdd(&st[5], s1);
    atomicMaxF(&st[6], mx0); atomicMaxF(&st[7], mx1); atomicMaxF(&st[8], mxa);
  }
}

__global__ void apply_norm_kernel(const float* __restrict__ x, const float* __restrict__ st,
                                  float* __restrict__ h) {
  int i = blockIdx.x * blockDim.x + threadIdx.x;
  if (i >= NN) return;
  float cx = x[i * 3 + 0], cy = x[i * 3 + 1], ar = x[i * 3 + 2];
  bool cond = (st[1] - st[3]) > (st[0] - st[2]);
  const float th = 1.5707963267948966f;
  float cth = cosf(th), sth = sinf(th);
  float r0 = cx * cth - cy * sth;
  float r1 = cx * sth + cy * cth;
  float a = cond ? r0 : cx;
  float b = cond ? r1 : cy;
  float m0 = st[4] / (float)NN, m1 = st[5] / (float)NN;
  h[i * 3 + 0] = (a - m0) / st[6];
  h[i * 3 + 1] = (b - m1) / st[7];
  h[i * 3 + 2] = ar / st[8];
}

// ---------- degree (computed once, shared across all 7 layers) ----------
__global__ void degree_kernel(const int* __restrict__ dst, float* __restrict__ deg) {
  int e = blockIdx.x * blockDim.x + threadIdx.x;
  if (e < NE) atomicAdd(&deg[dst[e]], 1.0f);
}
__global__ void invdeg_kernel(float* __restrict__ deg) {
  int i = blockIdx.x * blockDim.x + threadIdx.x;
  if (i < NN) deg[i] = 1.0f / fmaxf(deg[i], 1.0f);
}

// ---------- edge scatter: summed[dst] += x[src]  (one thread per edge x 4-feature chunk) ----------
__global__ void scatter_kernel(const float* __restrict__ xin, int fi,
                               const int* __restrict__ src, const int* __restrict__ dst,
                               float* __restrict__ summed) {
  int chunks = (fi + 3) >> 2;
  long long idx = (long long)blockIdx.x * blockDim.x + threadIdx.x;
  long long tot = (long long)NE * chunks;
  if (idx >= tot) return;
  int e = (int)(idx / chunks);
  int c = (int)(idx - (long long)e * chunks);
  int s = src[e], d = dst[e];
  int k0 = c << 2;
  int kn = (k0 + 4 < fi) ? 4 : (fi - k0);
  const float* xs = xin + (size_t)s * fi + k0;
  float* sd = summed + (size_t)d * fi + k0;
  for (int k = 0; k < kn; ++k) atomicAdd(&sd[k], xs[k]);
}

// ---------- pack A = [mean || x] with zero pad to K ----------
__global__ void pack_kernel(const float* __restrict__ summed, const float* __restrict__ xin,
                            const float* __restrict__ invdeg, float* __restrict__ Xaug,
                            int fi, int K) {
  long long idx = (long long)blockIdx.x * blockDim.x + threadIdx.x;
  long long tot = (long long)NN * K;
  if (idx >= tot) return;
  int node = (int)(idx / K);
  int k = (int)(idx - (long long)node * K);
  float v;
  if (k < fi)            v = summed[(size_t)node * fi + k] * invdeg[node];
  else if (k < 2 * fi)   v = xin[(size_t)node * fi + (k - fi)];
  else                   v = 0.0f;
  Xaug[idx] = v;
}

// ---------- fused WMMA GEMM:  out = tanh(A @ [wl;wr] + b) (+ res) ----------
// A: M x K (K mult of 4). Per block: B (K x Npad) staged in LDS cooperatively;
// each wave's contiguous 16xK A-tile staged via global_load_async_to_lds_b128
// (coalesced 512B/instr, ASYNCcnt-tracked), then WMMA fragments fed from LDS.
template <int NT>
__global__ void gemm_tanh_kernel(const float* __restrict__ A, int K,
                                 const float* __restrict__ wlp, int Kl,
                                 const float* __restrict__ wrp, int Kr,
                                 const float* __restrict__ bias,
                                 const float* __restrict__ res,
                                 float* __restrict__ out,
                                 int M, int fo) {
  constexpr int Npad = NT * 16;
  extern __shared__ float lds[];
  float* Bs = lds;               // K * Npad
  float* bs = Bs + K * Npad;     // Npad
  float* As = bs + Npad;         // 4 waves * 16 * K
  int t = threadIdx.x;
  int wave = t >> 5, lane = t & 31;
  int m0 = (blockIdx.x * 4 + wave) * 16;

  // issue async A-tile copy first (overlaps with cooperative B fill below)
  if (m0 < M) {
    const char* gbase = (const char*)(A + (size_t)m0 * K);     // 16*K floats, contiguous
    unsigned lbase = (unsigned)(uintptr_t)(As + wave * 16 * K);
    int chunks = 4 * K;                                        // 16-byte chunks in tile
    for (int c = lane; c < chunks; c += 32)
      async_copy_b128(lbase + (unsigned)c * 16u, gbase + (size_t)c * 16u);
  }

  // cooperative B + bias fill into LDS
  int totB = K * Npad;
  for (int idx = t; idx < totB; idx += blockDim.x) {
    int k = idx / Npad, n = idx - k * Npad;
    float v = 0.0f;
    if (n < fo) {
      if (k < Kl)             v = wlp[k * fo + n];
      else if (k < Kl + Kr)   v = wrp[(k - Kl) * fo + n];
    }
    Bs[idx] = v;
  }
  for (int n = t; n < Npad; n += blockDim.x) bs[n] = (n < fo) ? bias[n] : 0.0f;
  __syncthreads();

  if (m0 >= M) return;   // wave-uniform: EXEC stays all-ones for WMMA
  wait_asynccnt0();      // own A-tile resident in LDS

  int mr = lane & 15;             // A row within tile
  int koff = (lane >> 4) << 1;    // lanes 16-31 hold K+2,K+3
  int ncol = lane & 15;           // B/D column

  v8f acc[NT];
#pragma unroll
  for (int nt = 0; nt < NT; ++nt) {
    float bv = bs[nt * 16 + ncol];
#pragma unroll
    for (int v = 0; v < 8; ++v) acc[nt][v] = bv;  // bias folded into C
  }

  const float* Aw = As + wave * 16 * K + mr * K;
  for (int k0 = 0; k0 < K; k0 += 4) {
    v2f a;
    a.x = Aw[k0 + koff];
    a.y = Aw[k0 + koff + 1];
#pragma unroll
    for (int nt = 0; nt < NT; ++nt) {
      v2f b;
      b.x = Bs[(k0 + koff) * Npad + nt * 16 + ncol];
      b.y = Bs[(k0 + koff + 1) * Npad + nt * 16 + ncol];
      acc[nt] = __builtin_amdgcn_wmma_f32_16x16x4_f32(
          false, a, false, b, (short)0, acc[nt], false, false);
    }
  }

  int rbase = m0 + ((lane >> 4) << 3);  // D: vgpr v -> row v (+8 for upper half-wave)
#pragma unroll
  for (int nt = 0; nt < NT; ++nt) {
    int n = nt * 16 + ncol;
    if (n < fo) {
#pragma unroll
      for (int v = 0; v < 8; ++v) {
        int row = rbase + v;
        float val = tanhf(acc[nt][v]);
        if (res) val += res[(size_t)row * fo + n];  // residual is post-activation
        out[(size_t)row * fo + n] = val;
      }
    }
  }
}

// ---------- lin3 (16->2) + softmax ----------
__global__ void lin3_softmax_kernel(const float* __restrict__ h, const float* __restrict__ w,
                                    const float* __restrict__ b, float* __restrict__ out) {
  int i = blockIdx.x * blockDim.x + threadIdx.x;
  if (i >= NN) return;
  const float* hr = h + (size_t)i * 16;
  float z0 = b[0], z1 = b[1];
  for (int k = 0; k < 16; ++k) {
    float hv = hr[k];
    z0 += hv * w[k * 2 + 0];
    z1 += hv * w[k * 2 + 1];
  }
  float m = fmaxf(z0, z1);
  float e0 = expf(z0 - m), e1 = expf(z1 - m);
  float s = e0 + e1;
  out[(size_t)i * 2 + 0] = e0 / s;
  out[(size_t)i * 2 + 1] = e1 / s;
}

// ---------------------------------------------------------------------------
extern "C" void kernel_launch(void* const* d_in, const int* in_sizes, int n_in,
                              void* d_out, int out_size, void* d_ws, size_t ws_size,
                              hipStream_t stream) {
  (void)in_sizes; (void)n_in; (void)out_size; (void)ws_size;
  const float* x  = (const float*)d_in[0];
  const int* ei   = (const int*)d_in[1];
  const int* src  = ei;
  const int* dst  = ei + NE;
  const float *wl[7], *bl[7], *wr[7];
  for (int l = 0; l < 7; ++l) {
    wl[l] = (const float*)d_in[2 + 3 * l];
    bl[l] = (const float*)d_in[3 + 3 * l];
    wr[l] = (const float*)d_in[4 + 3 * l];
  }
  const float* l1w = (const float*)d_in[23]; const float* l1b = (const float*)d_in[24];
  const float* l2w = (const float*)d_in[25]; const float* l2b = (const float*)d_in[26];
  const float* l3w = (const float*)d_in[27]; const float* l3b = (const float*)d_in[28];

  // workspace carve (floats)
  float* W = (float*)d_ws;
  size_t o = 0;
  float* stats  = W + o; o += 16;
  float* deg    = W + o; o += NN;                 // becomes invdeg in place
  float* hnorm  = W + o; o += (size_t)NN * 3;
  float* x1     = W + o; o += (size_t)NN * 8;
  float* x2     = W + o; o += (size_t)NN * 16;
  float* x3     = W + o; o += (size_t)NN * 32;
  float* x4     = W + o; o += (size_t)NN * 64;
  float* hA     = W + o; o += (size_t)NN * 64;
  float* hB     = W + o; o += (size_t)NN * 64;
  float* summed = W + o; o += (size_t)NN * 64;
  float* Xaug   = W + o; o += (size_t)NN * 128;

  // ---- normalize ----
  stats_init_kernel<<<1, 64, 0, stream>>>(stats);
  reduce_minmax_kernel<<<512, 256, 0, stream>>>(x, stats);
  reduce_stats2_kernel<<<512, 256, 0, stream>>>(x, stats);
  apply_norm_kernel<<<(NN + 255) / 256, 256, 0, stream>>>(x, stats, hnorm);

  // ---- degree (once) ----
  zero_kernel<<<512, 256, 0, stream>>>(deg, NN);
  degree_kernel<<<(NE + 255) / 256, 256, 0, stream>>>(dst, deg);
  invdeg_kernel<<<(NN + 255) / 256, 256, 0, stream>>>(deg);

  auto launch_gemm = [&](const float* A, int K, const float* wlp, int Kl,
                         const float* wrp, int Kr, const float* bp,
                         const float* resp, float* outp, int fo) {
    int Npad = (fo + 15) & ~15;
    int tiles = (NN + 15) / 16;
    int blocks = (tiles + 3) / 4;
    size_t shmem = (size_t)(K * Npad + Npad + 64 * K) * sizeof(float);
    switch (Npad / 16) {
      case 1: gemm_tanh_kernel<1><<<blocks, 128, shmem, stream>>>(A, K, wlp, Kl, wrp, Kr, bp, resp, outp, NN, fo); break;
      case 2: gemm_tanh_kernel<2><<<blocks, 128, shmem, stream>>>(A, K, wlp, Kl, wrp, Kr, bp, resp, outp, NN, fo); break;
      default: gemm_tanh_kernel<4><<<blocks, 128, shmem, stream>>>(A, K, wlp, Kl, wrp, Kr, bp, resp, outp, NN, fo); break;
    }
  };

  auto run_sage = [&](const float* xin, int fi, int fo, int li,
                      const float* resp, float* outp) {
    int K = (2 * fi + 3) & ~3;
    zero_kernel<<<1024, 256, 0, stream>>>(summed, (long long)NN * fi);
    int chunks = (fi + 3) >> 2;
    long long swork = (long long)NE * chunks;
    scatter_kernel<<<(int)((swork + 255) / 256), 256, 0, stream>>>(xin, fi, src, dst, summed);
    long long pwork = (long long)NN * K;
    pack_kernel<<<(int)((pwork + 255) / 256), 256, 0, stream>>>(summed, xin, deg, Xaug, fi, K);
    launch_gemm(Xaug, K, wl[li], fi, wr[li], fi, bl[li], resp, outp, fo);
  };

  // ---- SAGE stack ----
  run_sage(hnorm, 3,  8,  0, nullptr, x1);
  run_sage(x1,    8,  16, 1, nullptr, x2);
  run_sage(x2,    16, 32, 2, nullptr, x3);
  run_sage(x3,    32, 64, 3, nullptr, x4);
  run_sage(x4,    64, 32, 4, x3,      hA);
  run_sage(hA,    32, 16, 5, x2,      hB);
  run_sage(hB,    16, 8,  6, x1,      hA);

  // ---- MLP head ----
  launch_gemm(hA, 8,  l1w, 8,  nullptr, 0, l1b, nullptr, hB, 32);
  launch_gemm(hB, 32, l2w, 32, nullptr, 0, l2b, nullptr, hA, 16);
  lin3_softmax_kernel<<<(NN + 255) / 256, 256, 0, stream>>>(hA, l3w, l3b, (float*)d_out);
}